// DynamicCodecSeparator_87282325389774
// MI455X (gfx1250) — compile-verified
//
#include <hip/hip_runtime.h>
#include <hip/hip_bf16.h>

typedef float v2f __attribute__((ext_vector_type(2)));
typedef float v8f __attribute__((ext_vector_type(8)));

#define EMB_D      512
#define PAD_D      516              // 512 + 4 floats padding -> conflict-free LDS banks
#define ROWS_PB    64               // rows per block (4 waves x 16-row tile)
#define CODE_TILE  16
#define NCODES     1024
#define NTILES     (NCODES / CODE_TILE)
#define BLOCK_T    128

#if __has_builtin(__builtin_amdgcn_global_load_async_to_lds_b128)
#define HAS_ASYNC_LDS 1
typedef int v4i_vs __attribute__((__vector_size__(4 * sizeof(int))));
typedef __attribute__((address_space(1))) v4i_vs as1_v4i;   // global (__device__) side
typedef __attribute__((address_space(3))) v4i_vs as3_v4i;   // LDS (__shared__) side
#endif

__device__ __forceinline__ void wait_async_zero() {
#if defined(HAS_ASYNC_LDS)
#if __has_builtin(__builtin_amdgcn_s_wait_asynccnt)
    __builtin_amdgcn_s_wait_asynccnt(0);
#else
    asm volatile("s_wait_asynccnt 0" ::: "memory");
#endif
#endif
}

// ---------------- Stage 1: codebook squared norms ----------------
__global__ void vq_norms_kernel(const float* __restrict__ E, float* __restrict__ norms) {
    const int k   = blockIdx.x;          // one code per block
    const int tid = threadIdx.x;         // 128 threads
    const float4* e4 = reinterpret_cast<const float4*>(E + (size_t)k * EMB_D);
    float4 v = e4[tid];                  // 128*4 = 512 floats
    float s = v.x * v.x + v.y * v.y + v.z * v.z + v.w * v.w;
    #pragma unroll
    for (int off = 16; off >= 1; off >>= 1) s += __shfl_xor(s, off, 32);
    __shared__ float ws[4];
    const int w = tid >> 5, l = tid & 31;
    if (l == 0) ws[w] = s;
    __syncthreads();
    if (tid == 0) norms[k] = ws[0] + ws[1] + ws[2] + ws[3];
}

// ---------------- Stage 2: fp32 WMMA distance + argmin ----------------
extern __shared__ __align__(16) float smem[];

// Issue async DMA of one 16x512 code tile into (padded) LDS buffer.
__device__ __forceinline__ void issue_tile(const float* __restrict__ E, int t,
                                           float* __restrict__ buf, int tid) {
#if defined(HAS_ASYNC_LDS)
    const float* g = E + (size_t)t * CODE_TILE * EMB_D + tid * 4;  // col = tid*4
    float*       l = buf + tid * 4;
    #pragma unroll
    for (int i = 0; i < CODE_TILE; ++i) {
        __builtin_amdgcn_global_load_async_to_lds_b128(
            (as1_v4i*)(g + i * EMB_D), (as3_v4i*)(l + i * PAD_D), 0, 0);
    }
#else
    const float4* E4 = reinterpret_cast<const float4*>(E) + (long)t * CODE_TILE * (EMB_D / 4);
    for (int i = tid; i < CODE_TILE * EMB_D / 4; i += BLOCK_T) {
        float4 v = E4[i];
        const int row = (i * 4) / EMB_D, col = (i * 4) % EMB_D;
        *reinterpret_cast<float4*>(&buf[row * PAD_D + col]) = v;
    }
#endif
}

__global__ __launch_bounds__(BLOCK_T, 1)
void vq_argmin_kernel(const float* __restrict__ X, const float* __restrict__ E,
                      const float* __restrict__ norms, float* __restrict__ idx_out) {
    float* As = smem;                              // [ROWS_PB][PAD_D]
    float* Bs = smem + ROWS_PB * PAD_D;            // [2][CODE_TILE][PAD_D] (ping-pong)
    float* nE = Bs + 2 * CODE_TILE * PAD_D;        // [NCODES]

    const int tid  = threadIdx.x;
    const int wave = tid >> 5;
    const int lane = tid & 31;
    const int half = lane >> 4;                    // 0: lanes 0-15, 1: lanes 16-31
    const int l16  = lane & 15;
    const long r0  = (long)blockIdx.x * ROWS_PB;

    // Stage input rows into LDS (coalesced float4, padded rows)
    const float4* X4 = reinterpret_cast<const float4*>(X) + r0 * (EMB_D / 4);
    for (int i = tid; i < ROWS_PB * EMB_D / 4; i += BLOCK_T) {
        float4 v = X4[i];
        const int row = (i * 4) / EMB_D, col = (i * 4) % EMB_D;
        *reinterpret_cast<float4*>(&As[row * PAD_D + col]) = v;
    }
    for (int i = tid; i < NCODES; i += BLOCK_T) nE[i] = norms[i];

    // Prefetch first code tile into buffer 0 (establishes ping-pong pipeline).
    issue_tile(E, 0, Bs, tid);

    // Per-lane fragment base pointers (A 16x4 / B 4x16 fp32 WMMA layouts):
    //   lanes 0-15 supply K = k0, k0+1 ; lanes 16-31 supply K = k0+2, k0+3
    const float* aptr  = As + (wave * 16 + l16) * PAD_D + 2 * half;
    const float* bbase = Bs + l16 * PAD_D + 2 * half;

    float minv[8];
    int   mini[8];
    #pragma unroll
    for (int j = 0; j < 8; ++j) { minv[j] = 3.4e38f; mini[j] = 0; }

    for (int t = 0; t < NTILES; ++t) {
        // Own-wave DMA for tile t done; barrier makes all waves' DMA visible
        // and guarantees everyone finished reading the other buffer.
        wait_async_zero();
        __syncthreads();
        if (t + 1 < NTILES)
            issue_tile(E, t + 1, Bs + ((t + 1) & 1) * CODE_TILE * PAD_D, tid);
#if !defined(HAS_ASYNC_LDS)
        __syncthreads();   // sync path: tile t+1 was written with plain stores
#endif
        const float* bptr = bbase + (t & 1) * CODE_TILE * PAD_D;

        // C[m][n] = sum_k X[m][k]*E[n][k]; two accumulators -> two independent
        // XDL dependency chains on v_wmma_f32_16x16x4_f32.
        v8f c0 = {0.f, 0.f, 0.f, 0.f, 0.f, 0.f, 0.f, 0.f};
        v8f c1 = {0.f, 0.f, 0.f, 0.f, 0.f, 0.f, 0.f, 0.f};
        #pragma unroll 8
        for (int k0 = 0; k0 < EMB_D; k0 += 8) {
            v2f a0 = *reinterpret_cast<const v2f*>(aptr + k0);
            v2f b0 = *reinterpret_cast<const v2f*>(bptr + k0);
            c0 = __builtin_amdgcn_wmma_f32_16x16x4_f32(
                     false, a0, false, b0, (short)0, c0, false, false);
            v2f a1 = *reinterpret_cast<const v2f*>(aptr + k0 + 4);
            v2f b1 = *reinterpret_cast<const v2f*>(bptr + k0 + 4);
            c1 = __builtin_amdgcn_wmma_f32_16x16x4_f32(
                     false, a1, false, b1, (short)0, c1, false, false);
        }

        // dist(m, code) = ||e||^2 - 2*dot ; argmin over this tile's 16 codes.
        // C layout: lane L, VGPR j -> (M = j + 8*(L>>4), N = L&15)
        const int   code = t * CODE_TILE + l16;
        const float ne   = nE[code];
        #pragma unroll
        for (int j = 0; j < 8; ++j) {
            const float vorig = ne - 2.0f * (c0[j] + c1[j]);
            float v = vorig;
            #pragma unroll
            for (int off = 8; off >= 1; off >>= 1)   // min butterfly per 16-lane half
                v = fminf(v, __shfl_xor(v, off, 32));
            // lowest-index winner via ballot (matches jnp.argmin tie-break)
            const unsigned mask = (unsigned)__ballot(vorig == v);
            const unsigned hm   = half ? (mask >> 16) : (mask & 0xffffu);
            const int ci = t * CODE_TILE + (__ffs(hm) - 1);
            if (v < minv[j]) { minv[j] = v; mini[j] = ci; }  // strict < keeps earliest
        }
    }

    if (l16 == 0) {  // lane 0 -> rows M=0..7, lane 16 -> rows M=8..15
        #pragma unroll
        for (int j = 0; j < 8; ++j) {
            const long row = r0 + wave * 16 + half * 8 + j;
            idx_out[row] = (float)mini[j];
        }
    }
}

// ---------------- Stage 3: gather + straight-through + loss partials ----------------
__global__ void vq_gather_kernel(const float* __restrict__ X, const float* __restrict__ E,
                                 const float* __restrict__ idxf, float* __restrict__ qout,
                                 float* __restrict__ partial) {
    const long row = blockIdx.x;
    const int  tid = threadIdx.x;              // 128 threads
    const int  code = (int)idxf[row];
    const float4* e4 = reinterpret_cast<const float4*>(E + (size_t)code * EMB_D);
    const float4* x4 = reinterpret_cast<const float4*>(X + row * EMB_D);
    float4*       o4 = reinterpret_cast<float4*>(qout + row * EMB_D);

    float4 q = e4[tid];
    float4 x = x4[tid];
    float4 d; d.x = q.x - x.x; d.y = q.y - x.y; d.z = q.z - x.z; d.w = q.w - x.w;
    float4 o; o.x = x.x + d.x; o.y = x.y + d.y; o.z = x.z + d.z; o.w = x.w + d.w;
    o4[tid] = o;

    float s = d.x * d.x + d.y * d.y + d.z * d.z + d.w * d.w;
    #pragma unroll
    for (int off = 16; off >= 1; off >>= 1) s += __shfl_xor(s, off, 32);
    __shared__ float ws[4];
    const int w = tid >> 5, l = tid & 31;
    if (l == 0) ws[w] = s;
    __syncthreads();
    if (tid == 0) partial[row] = ws[0] + ws[1] + ws[2] + ws[3];
}

// ---------------- Stage 4: deterministic loss reduction ----------------
__global__ void vq_loss_kernel(const float* __restrict__ partial, int n, float scale,
                               float* __restrict__ out) {
    __shared__ float sm[256];
    const int tid = threadIdx.x;
    float s = 0.0f;
    for (int i = tid; i < n; i += 256) s += partial[i];   // fixed per-thread order
    sm[tid] = s;
    __syncthreads();
    for (int off = 128; off >= 1; off >>= 1) {            // fixed tree order
        if (tid < off) sm[tid] += sm[tid + off];
        __syncthreads();
    }
    if (tid == 0) out[0] = sm[0] * scale;
}

// ---------------- Launch ----------------
extern "C" void kernel_launch(void* const* d_in, const int* in_sizes, int n_in,
                              void* d_out, int out_size, void* d_ws, size_t ws_size,
                              hipStream_t stream) {
    const float* X = (const float*)d_in[0];   // (8,4096,512) fp32
    const float* E = (const float*)d_in[1];   // (1024,512)   fp32
    const int Ntot = in_sizes[0];             // N * D
    const int N    = Ntot / EMB_D;            // 32768 rows

    float* norms   = (float*)d_ws;            // [NCODES]
    float* partial = norms + NCODES;          // [N]

    float* qout = (float*)d_out;              // [N*D]
    float* idxf = qout + (size_t)N * EMB_D;   // [N]
    float* loss = idxf + N;                   // [1]

    vq_norms_kernel<<<NCODES, BLOCK_T, 0, stream>>>(E, norms);

    const size_t smem_bytes =
        (size_t)(ROWS_PB * PAD_D + 2 * CODE_TILE * PAD_D + NCODES) * sizeof(float);
    (void)hipFuncSetAttribute((const void*)vq_argmin_kernel,
                              hipFuncAttributeMaxDynamicSharedMemorySize,
                              (int)smem_bytes);
    vq_argmin_kernel<<<N / ROWS_PB, BLOCK_T, smem_bytes, stream>>>(X, E, norms, idxf);

    vq_gather_kernel<<<N, BLOCK_T, 0, stream>>>(X, E, idxf, qout, partial);

    vq_loss_kernel<<<1, 256, 0, stream>>>(partial, N, 0.25f / (float)Ntot, loss);
}